// SageForNode_42880953484118
// MI455X (gfx1250) — compile-verified
//
#include <hip/hip_runtime.h>
#include <hip/hip_bf16.h>

#define N_NODES 100000
#define N_EDGES 1000000
#define F_IN    64
#define F_HID   64
#define F_OUT   32

typedef __attribute__((ext_vector_type(2))) float v2f;
typedef __attribute__((ext_vector_type(8))) float v8f;

// ---------------------------------------------------------------------------
// Utility: zero-fill a float buffer
// ---------------------------------------------------------------------------
__global__ void fill_zero_kernel(float* __restrict__ p, long long n) {
    long long i = (long long)blockIdx.x * blockDim.x + threadIdx.x;
    long long stride = (long long)gridDim.x * blockDim.x;
    for (; i < n; i += stride) p[i] = 0.0f;
}

// ---------------------------------------------------------------------------
// Degree: deg[dst] += 1 per edge
// ---------------------------------------------------------------------------
__global__ void degree_kernel(const int* __restrict__ ei, float* __restrict__ deg) {
    int e = blockIdx.x * blockDim.x + threadIdx.x;
    if (e < N_EDGES) {
        int dst = ei[N_EDGES + e];
        atomicAdd(&deg[dst], 1.0f);
    }
}

__global__ void invdeg_kernel(const float* __restrict__ deg, float* __restrict__ invdeg) {
    int i = blockIdx.x * blockDim.x + threadIdx.x;
    if (i < N_NODES) invdeg[i] = 1.0f / fmaxf(deg[i], 1.0f);
}

// ---------------------------------------------------------------------------
// Scatter: msg[dst][:] += feat[src][:]  (64 features, 16 groups of float4)
// One thread handles one (edge, 4-feature group): float4 gather + 4 f32 atomics.
// ---------------------------------------------------------------------------
__global__ void scatter_kernel(const float* __restrict__ feat,
                               const int* __restrict__ ei,
                               float* __restrict__ msg) {
    const long long total = (long long)N_EDGES * 16;
    long long i = (long long)blockIdx.x * blockDim.x + threadIdx.x;
    long long stride = (long long)gridDim.x * blockDim.x;
    for (; i < total; i += stride) {
        int e = (int)(i >> 4);
        int g = (int)(i & 15);
        int src = ei[e];
        int dst = ei[N_EDGES + e];
        const float4 v = *(const float4*)(feat + (size_t)src * 64 + g * 4);
        float* p = msg + (size_t)dst * 64 + g * 4;
        atomicAdd(p + 0, v.x);
        atomicAdd(p + 1, v.y);
        atomicAdd(p + 2, v.z);
        atomicAdd(p + 3, v.w);
    }
}

// ---------------------------------------------------------------------------
// Fused SAGE dense part via f32 WMMA:
//   out = act( (msg * invdeg) @ Wl + bias + xroot @ Wr )
// K = 64 (contraction dim), FOUT in {64, 32}.
// One block = 16 node rows; FOUT/16 waves, one 16x16 tile per wave.
// V_WMMA_F32_16X16X4_F32 fragment layouts (ISA 7.12.2):
//   A (16x4):  lane<16 -> M=lane, VGPR{0,1}=K{0,1}; lane>=16 -> K{2,3}
//   B (4x16):  lane half selects K pair; VGPR index selects K within pair
//   C/D:       VGPR v, lane<16 -> M=v, N=lane; lane>=16 -> M=8+v, N=lane-16
// ---------------------------------------------------------------------------
template <int FOUT, bool RELU>
__global__ void sage_wmma_kernel(const float* __restrict__ msg,
                                 const float* __restrict__ invdeg,
                                 const float* __restrict__ xroot,
                                 const float* __restrict__ Wl,
                                 const float* __restrict__ Wr,
                                 const float* __restrict__ bias,
                                 float* __restrict__ out) {
    constexpr int K   = 64;
    constexpr int PAD = 66;                 // avoid LDS bank conflicts
    __shared__ float Am[16 * PAD];          // mean tile
    __shared__ float Ax[16 * PAD];          // root-feature tile

    const int m0  = blockIdx.x * 16;
    const int tid = threadIdx.x;

    // Cooperative stage of the two 16x64 A tiles into LDS
    for (int i = tid; i < 16 * K; i += blockDim.x) {
        int r = i >> 6;
        int k = i & 63;
        float id = invdeg[m0 + r];
        Am[r * PAD + k] = msg[(size_t)(m0 + r) * K + k] * id;
        Ax[r * PAD + k] = xroot[(size_t)(m0 + r) * K + k];
    }
    __syncthreads();

    const int wave = tid >> 5;
    const int lane = tid & 31;
    const int half = lane >> 4;
    const int lrow = lane & 15;
    const int n0   = wave * 16;

    v8f acc = {};
#pragma unroll
    for (int kb = 0; kb < K; kb += 4) {
        const int ka = kb + 2 * half;       // this half-wave's K pair
        v2f a, b;
        // mean @ Wl
        a.x = Am[lrow * PAD + ka];
        a.y = Am[lrow * PAD + ka + 1];
        b.x = Wl[(size_t)ka * FOUT + n0 + lrow];
        b.y = Wl[(size_t)(ka + 1) * FOUT + n0 + lrow];
        acc = __builtin_amdgcn_wmma_f32_16x16x4_f32(false, a, false, b,
                                                    (short)0, acc, false, false);
        // xroot @ Wr
        a.x = Ax[lrow * PAD + ka];
        a.y = Ax[lrow * PAD + ka + 1];
        b.x = Wr[(size_t)ka * FOUT + n0 + lrow];
        b.y = Wr[(size_t)(ka + 1) * FOUT + n0 + lrow];
        acc = __builtin_amdgcn_wmma_f32_16x16x4_f32(false, a, false, b,
                                                    (short)0, acc, false, false);
    }

    const float bv = bias[n0 + lrow];
#pragma unroll
    for (int v = 0; v < 8; ++v) {
        float val = acc[v] + bv;
        if (RELU) val = fmaxf(val, 0.0f);
        out[(size_t)(m0 + v + 8 * half) * FOUT + n0 + lrow] = val;
    }
}

// ---------------------------------------------------------------------------
// Launch
// ---------------------------------------------------------------------------
extern "C" void kernel_launch(void* const* d_in, const int* in_sizes, int n_in,
                              void* d_out, int out_size, void* d_ws, size_t ws_size,
                              hipStream_t stream) {
    const float* x   = (const float*)d_in[0];
    const int*   ei  = (const int*)d_in[1];
    const float* W1l = (const float*)d_in[2];
    const float* b1  = (const float*)d_in[3];
    const float* W1r = (const float*)d_in[4];
    const float* W2l = (const float*)d_in[5];
    const float* b2  = (const float*)d_in[6];
    const float* W2r = (const float*)d_in[7];
    float* out = (float*)d_out;

    float* ws     = (float*)d_ws;
    float* msg    = ws;                                  // [N_NODES * 64]
    float* h      = msg + (size_t)N_NODES * 64;          // [N_NODES * 64]
    float* deg    = h   + (size_t)N_NODES * 64;          // [N_NODES]
    float* invdeg = deg + N_NODES;                       // [N_NODES]

    const long long msg_elems = (long long)N_NODES * 64;
    const int zb = 1024;

    // --- degree (shared by both layers) ---
    fill_zero_kernel<<<(N_NODES + zb * 4 - 1) / (zb * 4), zb, 0, stream>>>(deg, N_NODES);
    degree_kernel<<<(N_EDGES + 255) / 256, 256, 0, stream>>>(ei, deg);
    invdeg_kernel<<<(N_NODES + 255) / 256, 256, 0, stream>>>(deg, invdeg);

    // --- layer 1: scatter-mean + WMMA dense + ReLU ---
    fill_zero_kernel<<<4096, zb, 0, stream>>>(msg, msg_elems);
    {
        long long total = (long long)N_EDGES * 16;
        int blocks = (int)((total + 255) / 256);
        scatter_kernel<<<blocks, 256, 0, stream>>>(x, ei, msg);
    }
    sage_wmma_kernel<F_HID, true><<<N_NODES / 16, (F_HID / 16) * 32, 0, stream>>>(
        msg, invdeg, x, W1l, W1r, b1, h);

    // --- layer 2: scatter-mean + WMMA dense (no activation) ---
    fill_zero_kernel<<<4096, zb, 0, stream>>>(msg, msg_elems);
    {
        long long total = (long long)N_EDGES * 16;
        int blocks = (int)((total + 255) / 256);
        scatter_kernel<<<blocks, 256, 0, stream>>>(h, ei, msg);
    }
    sage_wmma_kernel<F_OUT, false><<<N_NODES / 16, (F_OUT / 16) * 32, 0, stream>>>(
        msg, invdeg, h, W2l, W2r, b2, out);
}